// RayTracer_3307124818577
// MI455X (gfx1250) — compile-verified
//
#include <hip/hip_runtime.h>
#include <stdint.h>

// MI455X (gfx1250, wave32) fused sphere-trace + sample + bisect kernel.
//
// Roofline: ~16 MB total traffic -> ~0.7 us @ 23.3 TB/s; <=0.5 G VALU ops
// (sqrt/TRANS bound) -> also us-scale. Optimal form: single fused per-ray
// pass, one read per input, one write per output, wave32 predication.
// No WMMA: the reference contains no matrix contraction; the dense sampler
// is an early-exit first-sign-change scan that a 16x16 tile formulation
// would make slower (compaction + cross-lane argmin > 2 FMA/elem saved).
// CDNA5-specific paths used: GLOBAL_LOAD_ASYNC_TO_LDS_* (+ s_wait_asynccnt)
// to overlap scalar-param fetch with the b96 ray loads, and
// global_prefetch_b8 on the ray vectors.

#define SDF_TH 5e-5f
#define RADIUS 0.8f
#define TRACE_ITERS 16
#define N_STEPS 128
#define BLOCK 256

#define AS1 __attribute__((address_space(1)))
#define AS3 __attribute__((address_space(3)))

#if __has_builtin(__builtin_amdgcn_global_load_async_to_lds_b32)
#define HAVE_ASYNC_B32 1
#else
#define HAVE_ASYNC_B32 0
#endif
#if __has_builtin(__builtin_amdgcn_global_load_async_to_lds_b8)
#define HAVE_ASYNC_B8 1
#else
#define HAVE_ASYNC_B8 0
#endif

__device__ __forceinline__ float sdf_at(float px, float py, float pz,
                                        float cx, float cy, float cz) {
  float ax = px - cx, ay = py - cy, az = pz - cz;
  return sqrtf(fmaf(ax, ax, fmaf(ay, ay, az * az))) - RADIUS;
}

__global__ __launch_bounds__(BLOCK) void ray_trace_fused(
    const float* __restrict__ ray_o, const float* __restrict__ ray_d,
    const float* __restrict__ min_dis, const float* __restrict__ max_dis,
    const float* __restrict__ center, const uint8_t* __restrict__ work_mask,
    float* __restrict__ out, int n) {
#if HAVE_ASYNC_B32
  __shared__ float s_mn[BLOCK];
  __shared__ float s_mx[BLOCK];
#endif
#if HAVE_ASYNC_B8
  __shared__ uint8_t s_wm[BLOCK];
#endif

  int i = blockIdx.x * blockDim.x + threadIdx.x;
  int ci = (i < n) ? i : (n - 1);  // clamp so all lanes issue valid DMA
  size_t c3 = 3 * (size_t)ci;

  // 1) Kick off async global->LDS DMA of the scalar per-ray params.
#if HAVE_ASYNC_B32
  __builtin_amdgcn_global_load_async_to_lds_b32(
      (AS1 int*)(min_dis + ci), (AS3 int*)(&s_mn[threadIdx.x]), 0, 0);
  __builtin_amdgcn_global_load_async_to_lds_b32(
      (AS1 int*)(max_dis + ci), (AS3 int*)(&s_mx[threadIdx.x]), 0, 0);
#endif
#if HAVE_ASYNC_B8
  __builtin_amdgcn_global_load_async_to_lds_b8(
      (AS1 char*)(work_mask + ci), (AS3 char*)(&s_wm[threadIdx.x]), 0, 0);
#endif

  // 2) Prefetch the ray vectors while the DMA is in flight.
  __builtin_prefetch(ray_o + c3, 0, 3);
  __builtin_prefetch(ray_d + c3, 0, 3);

  if (i >= n) return;
  size_t i3 = 3 * (size_t)i;

  // 3) Direct coalesced loads (b96) + uniform scalar loads, overlapping DMA.
  float cx = center[0], cy = center[1], cz = center[2];
  float ox = ray_o[i3 + 0], oy = ray_o[i3 + 1], oz = ray_o[i3 + 2];
  float dx = ray_d[i3 + 0], dy = ray_d[i3 + 1], dz = ray_d[i3 + 2];

  // 4) Consume the DMA results (per-wave ASYNCcnt ordering; each lane reads
  //    only its own slot, so no block barrier is needed).
#if HAVE_ASYNC_B32
#if __has_builtin(__builtin_amdgcn_s_wait_asynccnt)
  __builtin_amdgcn_s_wait_asynccnt(0);
#else
  asm volatile("s_wait_asynccnt 0x0");
#endif
  asm volatile("" ::: "memory");  // keep LDS reads below the wait
  float mn = s_mn[threadIdx.x];
  float mx = s_mx[threadIdx.x];
#else
  float mn = min_dis[i];
  float mx = max_dis[i];
#endif
#if HAVE_ASYNC_B8
  bool work = s_wm[threadIdx.x] != 0;
#else
  bool work = work_mask[i] != 0;  // jnp bool_ = 1 byte per element
#endif

  // ---------------- sphere tracing ----------------
  float acc = mn;
  float px = fmaf(dx, acc, ox), py = fmaf(dy, acc, oy), pz = fmaf(dz, acc, oz);
  float f = sdf_at(px, py, pz, cx, cy, cz);
  bool unfinished = work;
  for (int it = 0; it < TRACE_ITERS; ++it) {
    unfinished = unfinished && (fabsf(f) > SDF_TH) && (acc < mx);
    if (__ballot((int)unfinished) == 0ull) break;  // wave32 early exit
    float step = unfinished ? f : 0.0f;
    acc += step;
    px = fmaf(dx, step, px);
    py = fmaf(dy, step, py);
    pz = fmaf(dz, step, pz);
    float fn = sdf_at(px, py, pz, cx, cy, cz);
    f = unfinished ? fn : f;
  }
  unfinished = unfinished && (fabsf(f) > SDF_TH) && (acc < mx);
  bool convergent = work && !unfinished && (fabsf(f) <= SDF_TH) && (acc < mx);

  float o_conv = convergent ? 1.0f : 0.0f;
  float opx = px, opy = py, opz = pz, of = f, oacc = acc;

  // ------- dense sampling + bisection for unfinished rays -------
  if (unfinished) {
    float smin = (f > 0.0f) ? acc : mn;
    float smax = (f > 0.0f) ? mx : acc;
    float range = smax - smin;

    // argmin(sign(sdf_i)*(N_STEPS-i)) with first-min tie break is exactly
    // "first i with sdf_i < 0"; no negative sample => root_mask = false.
    const float tstep = 1.0f / (float)(N_STEPS - 1);  // linspace delta
    bool found = false;
    int idx = 0;
    float zh = 0.0f, prev_z = 0.0f, prev_f = 0.0f;
    for (int s = 0; s < N_STEPS; ++s) {
      float t = (float)s * tstep;
      float z = fmaf(t, range, smin);
      float sx = fmaf(dx, z, ox), sy = fmaf(dy, z, oy), sz = fmaf(dz, z, oz);
      float fs = sdf_at(sx, sy, sz, cx, cy, cz);
      if (fs < 0.0f) {  // global minimum of sign*weight: stop here
        found = true;
        idx = s;
        zh = z;
        break;
      }
      prev_z = z;
      prev_f = fs;
    }

    bool root = found && (idx >= 1);
    if (root) {
      // bracket: [prev (>=0), current (<0)]
      float d_lo = prev_z, d_hi = zh;
      float d_mid = 0.5f * (d_lo + d_hi);
      bool bw = (prev_f > 0.0f);  // f_high < 0 guaranteed
      for (int it = 0; it < 64 && bw; ++it) {
        float qx = fmaf(dx, d_mid, ox), qy = fmaf(dy, d_mid, oy),
              qz = fmaf(dz, d_mid, oz);
        float fm = sdf_at(qx, qy, qz, cx, cy, cz);
        if (fm > 0.0f)
          d_lo = d_mid;
        else
          d_hi = d_mid;
        d_mid = 0.5f * (d_lo + d_hi);
        bw = (d_hi - d_lo) > 2.0f * SDF_TH;
      }
      float qx = fmaf(dx, d_mid, ox), qy = fmaf(dy, d_mid, oy),
            qz = fmaf(dz, d_mid, oz);
      o_conv = 1.0f;
      opx = qx;
      opy = qy;
      opz = qz;
      of = sdf_at(qx, qy, qz, cx, cy, cz);
      oacc = d_mid;
    } else {
      o_conv = 0.0f;
      opx = opy = opz = 0.0f;
      of = 0.0f;
      oacc = 0.0f;
    }
  }

  // Output layout: [conv N][pts 3N][f N][acc N] (floats; conv as 0/1)
  out[i] = o_conv;
  float* pts_out = out + n;
  pts_out[i3 + 0] = opx;
  pts_out[i3 + 1] = opy;
  pts_out[i3 + 2] = opz;
  out[(size_t)4 * n + i] = of;
  out[(size_t)5 * n + i] = oacc;
}

extern "C" void kernel_launch(void* const* d_in, const int* in_sizes, int n_in,
                              void* d_out, int out_size, void* d_ws,
                              size_t ws_size, hipStream_t stream) {
  const float* ray_o = (const float*)d_in[0];
  const float* ray_d = (const float*)d_in[1];
  const float* min_dis = (const float*)d_in[2];
  const float* max_dis = (const float*)d_in[3];
  const float* center = (const float*)d_in[4];
  const uint8_t* work_mask = (const uint8_t*)d_in[5];
  float* out = (float*)d_out;

  int n = in_sizes[2];  // min_dis element count == ray count
  int grid = (n + BLOCK - 1) / BLOCK;
  ray_trace_fused<<<grid, BLOCK, 0, stream>>>(ray_o, ray_d, min_dis, max_dis,
                                              center, work_mask, out, n);
  (void)n_in;
  (void)out_size;
  (void)d_ws;
  (void)ws_size;
}